// Describe_4398046511580
// MI455X (gfx1250) — compile-verified
//
#include <hip/hip_runtime.h>
#include <hip/hip_bf16.h>

// Problem sizes (compile-time constants from the reference)
#define B_N   256
#define C_N   512
#define HW_N  196
#define ND_N  32
#define NA_N  2000

typedef float v2f __attribute__((ext_vector_type(2)));
typedef float v8f __attribute__((ext_vector_type(8)));

// ---------------------------------------------------------------------------
// Stage 1: attended[b][c] = (1/HW) * sum_hw mask[b][hw] * features[b][c][hw]
// One block per batch sample. Mask cached in LDS. One wave32 per channel,
// lanes stride over HW, shuffle reduction (wave32 on gfx1250).
// ---------------------------------------------------------------------------
__global__ void k_attend(const float* __restrict__ mask,
                         const float* __restrict__ feat,
                         float* __restrict__ attended) {
    const int b   = blockIdx.x;
    const int tid = threadIdx.x;           // 256 threads = 8 waves
    __shared__ float m_s[HW_N];
    if (tid < HW_N) m_s[tid] = mask[b * HW_N + tid];
    __syncthreads();

    const int wave = tid >> 5;
    const int lane = tid & 31;
    const float* fb = feat + (size_t)b * C_N * HW_N;

    for (int c = wave; c < C_N; c += 8) {
        const float* fc = fb + (size_t)c * HW_N;
        float s = 0.0f;
        for (int hw = lane; hw < HW_N; hw += 32)
            s += m_s[hw] * fc[hw];
        // wave32 butterfly reduction
        #pragma unroll
        for (int off = 16; off > 0; off >>= 1)
            s += __shfl_xor(s, off, 32);
        if (lane == 0)
            attended[b * C_N + c] = s * (1.0f / (float)HW_N);
    }
}

// ---------------------------------------------------------------------------
// Stage 2a: group batch samples by expert id (instance). Single block.
// counts[d] = #samples with instance==d; glist[d*256 + slot] = b (0-padded).
// Output of the whole pipeline is invariant to slot order.
// ---------------------------------------------------------------------------
__global__ void k_group(const int* __restrict__ instance,
                        int* __restrict__ counts,
                        int* __restrict__ glist) {
    __shared__ int cnt[ND_N];
    const int tid = threadIdx.x;          // 256 threads
    if (tid < ND_N) cnt[tid] = 0;
    for (int i = tid; i < ND_N * B_N; i += B_N) glist[i] = 0;  // pad with row 0
    __syncthreads();
    const int d    = instance[tid];
    const int slot = atomicAdd(&cnt[d], 1);
    glist[d * B_N + slot] = tid;
    __syncthreads();
    if (tid < ND_N) counts[tid] = cnt[tid];
}

// ---------------------------------------------------------------------------
// Stage 2b: grouped GEMM via V_WMMA_F32_16X16X4_F32 (full fp32 precision).
// Grid: x = d*16 + mtile  (32 experts x up to 16 row-tiles)
//       y = a-tile        (2000/16 = 125)
// One wave (32 threads) per 16x16 output tile, K = 512 in steps of 4.
//
// fp32 A 16x4 layout:  lane l holds A[M = l%16, K = 2*(l/16) + j] in VGPR j
// fp32 B 4x16 layout:  lane l holds B[K = 2*(l/16) + j, N = l%16] in VGPR j
// fp32 D 16x16 layout: lane l holds D[M = v + 8*(l/16), N = l%16] in VGPR v
// ---------------------------------------------------------------------------
__global__ void k_gemm(const float* __restrict__ attended,
                       const float* __restrict__ weight,
                       const float* __restrict__ bias,
                       const int* __restrict__ counts,
                       const int* __restrict__ glist,
                       float* __restrict__ out) {
    const int d     = blockIdx.x >> 4;
    const int mtile = blockIdx.x & 15;
    const int count = counts[d];
    if (mtile * 16 >= count) return;             // wave-uniform: EXEC stays all-1
    const int rowsInTile = min(16, count - mtile * 16);
    const int a0 = blockIdx.y * 16;

    __shared__ int rows_s[16];
    const int lane = threadIdx.x;                // blockDim.x == 32 (one wave)
    if (lane < 16) rows_s[lane] = glist[d * B_N + mtile * 16 + lane];
    __syncthreads();

    const int half = lane >> 4;                  // 0: lanes 0-15, 1: lanes 16-31
    const int n    = lane & 15;

    const int rowA = rows_s[n];                  // A-matrix row for this lane
    const float* arow = attended + (size_t)rowA * C_N + 2 * half;
    const float* wcol = weight + (size_t)d * C_N * NA_N + a0 + n;

    v8f acc = {};
    #pragma unroll 4
    for (int k0 = 0; k0 < C_N; k0 += 4) {
        v2f afrag = *(const v2f*)(arow + k0);    // K = k0+2h, k0+2h+1
        const int kb = k0 + 2 * half;
        v2f bfrag;
        bfrag.x = wcol[(size_t)kb * NA_N];       // B[K=kb,   N=a0+n]
        bfrag.y = wcol[(size_t)(kb + 1) * NA_N]; // B[K=kb+1, N=a0+n]
        acc = __builtin_amdgcn_wmma_f32_16x16x4_f32(
            /*neg_a=*/false, afrag, /*neg_b=*/false, bfrag,
            /*c_mod=*/(short)0, acc, /*reuse_a=*/false, /*reuse_b=*/false);
    }

    const float bia = bias[d * NA_N + a0 + n];
    #pragma unroll
    for (int v = 0; v < 8; ++v) {
        const int m = v + 8 * half;              // D row held by this lane/VGPR
        if (m < rowsInTile) {
            const int bidx = rows_s[m];
            out[(size_t)bidx * NA_N + a0 + n] = acc[v] + bia;
        }
    }
}

// ---------------------------------------------------------------------------
extern "C" void kernel_launch(void* const* d_in, const int* in_sizes, int n_in,
                              void* d_out, int out_size, void* d_ws, size_t ws_size,
                              hipStream_t stream) {
    const float* mask     = (const float*)d_in[0];   // [B,1,H,W]
    const float* features = (const float*)d_in[1];   // [B,C,H,W]
    const float* weight   = (const float*)d_in[2];   // [ND,C,NA]
    const float* bias     = (const float*)d_in[3];   // [ND,NA]
    const int*   instance = (const int*)d_in[4];     // [B] int32
    float*       out      = (float*)d_out;           // [B,NA]

    // Workspace layout
    float* attended = (float*)d_ws;                                   // B*C floats
    int*   counts   = (int*)((char*)d_ws + (size_t)B_N * C_N * 4);    // ND ints
    int*   glist    = counts + ND_N;                                  // ND*B ints

    k_attend<<<dim3(B_N), dim3(256), 0, stream>>>(mask, features, attended);
    k_group <<<dim3(1),   dim3(B_N), 0, stream>>>(instance, counts, glist);
    k_gemm  <<<dim3(ND_N * 16, NA_N / 16), dim3(32), 0, stream>>>(
        attended, weight, bias, counts, glist, out);
}